// PaDiM_23940147708586
// MI455X (gfx1250) — compile-verified
//
#include <hip/hip_runtime.h>
#include <math.h>

typedef __attribute__((ext_vector_type(2))) float v2f;
typedef __attribute__((ext_vector_type(8))) float v8f;

#define NB   16      // batch
#define NC   256     // channels
#define NP   3136    // pixels (56*56)
#define HH   56
#define WW   56
#define OUT  224
#define RAD  16      // gaussian radius (sigma=4, truncate=4)
#define SIGMA_F 4.0f

#define MWAVES 2     // waves (=pixels) per block in the WMMA kernel -> 32KB LDS
#define RBLK  512    // reduction blocks for min/max

// -------------------------------------------------------------------------
// Stage 1: Mahalanobis distance via V_WMMA_F32_16X16X4_F32.
// One wave handles one pixel p. delta[16 x 256] staged in LDS (A operand),
// inv_cov[p] (256x256) streamed from HBM as the B operand (read once).
// d2[b] = sum_d tmp[b,d]*delta[b,d], tmp = delta @ inv_cov[p].
// -------------------------------------------------------------------------
__global__ __launch_bounds__(32 * MWAVES)
void mahal_wmma_kernel(const float* __restrict__ emb,     // [NB, NC, NP]
                       const float* __restrict__ means,   // [NP, NC]
                       const float* __restrict__ inv_cov, // [NP, NC, NC]
                       float* __restrict__ dist)          // [NB, NP]
{
    __shared__ float sdelta[MWAVES][NB * NC];
    const int wave = threadIdx.x >> 5;
    const int lane = threadIdx.x & 31;
    const int p = blockIdx.x * MWAVES + wave;
    float* dl = sdelta[wave];

    // stage delta[b][c] = emb[b,c,p] - means[p,c] into LDS
    for (int i = lane; i < NB * NC; i += 32) {
        const int b = i >> 8;          // / NC
        const int c = i & (NC - 1);
        dl[i] = emb[(size_t)b * (NC * NP) + (size_t)c * NP + p]
              - means[(size_t)p * NC + c];
    }
    __syncthreads();

    const float* __restrict__ invp = inv_cov + (size_t)p * NC * NC;
    const int n  = lane & 15;          // N index within tile; also A-matrix M=lane&15
    const int kh = lane >> 4;          // 0: lanes 0-15 (K=0,1), 1: lanes 16-31 (K=2,3)

    float s[8];
    #pragma unroll
    for (int v = 0; v < 8; ++v) s[v] = 0.f;

    for (int dt = 0; dt < 16; ++dt) {  // 16 N-tiles of the 256 d-columns
        const int d = dt * 16 + n;
        v8f acc = {0.f, 0.f, 0.f, 0.f, 0.f, 0.f, 0.f, 0.f};
        #pragma unroll 4
        for (int c = 0; c < NC; c += 4) {       // K in chunks of 4 (f32 WMMA)
            const int k0 = c + (kh << 1);
            v2f a, bm;
            // A 16x4 layout: lane m=n, VGPR0=K(k0), VGPR1=K(k0+1)
            a.x  = dl[(n << 8) + k0];
            a.y  = dl[(n << 8) + k0 + 1];
            // B 4x16 layout: lane n=col, VGPR0=K(k0), VGPR1=K(k0+1)
            bm.x = invp[(size_t)k0 * NC + d];
            bm.y = invp[(size_t)(k0 + 1) * NC + d];
            acc = __builtin_amdgcn_wmma_f32_16x16x4_f32(
                false, a, false, bm, (short)0, acc, false, false);
        }
        // C/D layout: VGPR v holds row m = v + 8*kh, column n = lane&15.
        // Fold this tile into the quadratic form immediately.
        #pragma unroll
        for (int v = 0; v < 8; ++v) {
            const int m = v + (kh << 3);
            s[v] += acc[v] * dl[(m << 8) + d];
        }
    }

    // reduce across the 16 lanes of each half (offsets 1,2,4,8 stay in-half)
    #pragma unroll
    for (int off = 8; off >= 1; off >>= 1)
        #pragma unroll
        for (int v = 0; v < 8; ++v)
            s[v] += __shfl_xor(s[v], off, 32);

    if (n == 0) {   // lanes 0 (b=0..7) and 16 (b=8..15)
        #pragma unroll
        for (int v = 0; v < 8; ++v) {
            const int b = v + (kh << 3);
            dist[(size_t)b * NP + p] = sqrtf(s[v]);
        }
    }
}

// -------------------------------------------------------------------------
// Stage 2: bilinear 4x upsample, align_corners=False (half-pixel centers)
// -------------------------------------------------------------------------
__global__ void resize_bilinear_kernel(const float* __restrict__ in,  // [NB,56,56]
                                       float* __restrict__ out)       // [NB,224,224]
{
    const int idx = blockIdx.x * blockDim.x + threadIdx.x;
    const int total = NB * OUT * OUT;
    if (idx >= total) return;
    const int x = idx % OUT;
    const int y = (idx / OUT) % OUT;
    const int b = idx / (OUT * OUT);

    const float sx = (x + 0.5f) * 0.25f - 0.5f;
    const float sy = (y + 0.5f) * 0.25f - 0.5f;
    const int x0 = (int)floorf(sx);
    const int y0 = (int)floorf(sy);
    const float fx = sx - (float)x0;
    const float fy = sy - (float)y0;
    const int x0c = min(max(x0, 0), WW - 1);
    const int x1c = min(max(x0 + 1, 0), WW - 1);
    const int y0c = min(max(y0, 0), HH - 1);
    const int y1c = min(max(y0 + 1, 0), HH - 1);

    const float* inb = in + (size_t)b * HH * WW;
    const float v00 = inb[y0c * WW + x0c];
    const float v01 = inb[y0c * WW + x1c];
    const float v10 = inb[y1c * WW + x0c];
    const float v11 = inb[y1c * WW + x1c];
    const float top = v00 + fx * (v01 - v00);
    const float bot = v10 + fx * (v11 - v10);
    out[idx] = top + fy * (bot - top);
}

// -------------------------------------------------------------------------
// Stage 3/4: separable 33-tap gaussian, symmetric-reflect boundary
// -------------------------------------------------------------------------
__device__ __forceinline__ int reflect_idx(int i, int nmax) {
    if (i < 0) return -i - 1;
    if (i >= nmax) return 2 * nmax - 1 - i;
    return i;
}

__global__ void blur_h_kernel(const float* __restrict__ in, float* __restrict__ out)
{
    const int idx = blockIdx.x * blockDim.x + threadIdx.x;
    const int total = NB * OUT * OUT;
    if (idx >= total) return;
    const int x = idx % OUT;
    const int y = (idx / OUT) % OUT;
    const int b = idx / (OUT * OUT);
    const float* inb = in + (size_t)b * OUT * OUT;

    float wsum = 0.f, acc = 0.f;
    #pragma unroll
    for (int j = 0; j < 2 * RAD + 1; ++j) {
        const float t = (float)(j - RAD) / SIGMA_F;
        const float w = expf(-0.5f * t * t);
        wsum += w;
        const int yy = reflect_idx(y - RAD + j, OUT);
        acc += w * inb[yy * OUT + x];
    }
    out[idx] = acc / wsum;
}

__global__ void blur_w_kernel(const float* __restrict__ in, float* __restrict__ out)
{
    const int idx = blockIdx.x * blockDim.x + threadIdx.x;
    const int total = NB * OUT * OUT;
    if (idx >= total) return;
    const int x = idx % OUT;
    const int y = (idx / OUT) % OUT;
    const int b = idx / (OUT * OUT);
    const float* row = in + (size_t)b * OUT * OUT + (size_t)y * OUT;

    float wsum = 0.f, acc = 0.f;
    #pragma unroll
    for (int j = 0; j < 2 * RAD + 1; ++j) {
        const float t = (float)(j - RAD) / SIGMA_F;
        const float w = expf(-0.5f * t * t);
        wsum += w;
        const int xx = reflect_idx(x - RAD + j, OUT);
        acc += w * row[xx];
    }
    out[idx] = acc / wsum;
}

// -------------------------------------------------------------------------
// Stage 5: global min/max -> normalize (deterministic two-level reduction)
// -------------------------------------------------------------------------
__global__ void minmax_partial_kernel(const float* __restrict__ data, int n,
                                      float* __restrict__ partial)
{
    __shared__ float smin[256], smax[256];
    float mn = 3.4e38f, mx = -3.4e38f;
    for (int i = blockIdx.x * blockDim.x + threadIdx.x; i < n;
         i += gridDim.x * blockDim.x) {
        const float v = data[i];
        mn = fminf(mn, v);
        mx = fmaxf(mx, v);
    }
    smin[threadIdx.x] = mn;
    smax[threadIdx.x] = mx;
    __syncthreads();
    for (int s = 128; s > 0; s >>= 1) {
        if (threadIdx.x < s) {
            smin[threadIdx.x] = fminf(smin[threadIdx.x], smin[threadIdx.x + s]);
            smax[threadIdx.x] = fmaxf(smax[threadIdx.x], smax[threadIdx.x + s]);
        }
        __syncthreads();
    }
    if (threadIdx.x == 0) {
        partial[blockIdx.x * 2 + 0] = smin[0];
        partial[blockIdx.x * 2 + 1] = smax[0];
    }
}

__global__ void minmax_final_kernel(const float* __restrict__ partial,
                                    float* __restrict__ mm, int nblk)
{
    __shared__ float smin[256], smax[256];
    float mn = 3.4e38f, mx = -3.4e38f;
    for (int i = threadIdx.x; i < nblk; i += blockDim.x) {
        mn = fminf(mn, partial[i * 2 + 0]);
        mx = fmaxf(mx, partial[i * 2 + 1]);
    }
    smin[threadIdx.x] = mn;
    smax[threadIdx.x] = mx;
    __syncthreads();
    for (int s = 128; s > 0; s >>= 1) {
        if (threadIdx.x < s) {
            smin[threadIdx.x] = fminf(smin[threadIdx.x], smin[threadIdx.x + s]);
            smax[threadIdx.x] = fmaxf(smax[threadIdx.x], smax[threadIdx.x + s]);
        }
        __syncthreads();
    }
    if (threadIdx.x == 0) {
        mm[0] = smin[0];
        mm[1] = smax[0];
    }
}

__global__ void normalize_kernel(float* __restrict__ data,
                                 const float* __restrict__ mm, int n)
{
    const int i = blockIdx.x * blockDim.x + threadIdx.x;
    if (i >= n) return;
    const float mn = mm[0];
    const float inv = 1.0f / (mm[1] - mn);
    data[i] = (data[i] - mn) * inv;
}

// -------------------------------------------------------------------------
extern "C" void kernel_launch(void* const* d_in, const int* in_sizes, int n_in,
                              void* d_out, int out_size, void* d_ws, size_t ws_size,
                              hipStream_t stream)
{
    const float* emb    = (const float*)d_in[0];  // [16,256,3136]
    const float* means  = (const float*)d_in[1];  // [3136,256]
    const float* invcov = (const float*)d_in[2];  // [3136,256,256]
    // d_in[3..5] = h, w, crop_size scalars (compile-time constants here)

    float* ws      = (float*)d_ws;
    float* dist    = ws;                      // NB*NP        = 50176 f
    float* up      = dist + NB * NP;          // NB*OUT*OUT   = 802816 f
    float* tmp     = up + NB * OUT * OUT;     // NB*OUT*OUT   = 802816 f
    float* partial = tmp + NB * OUT * OUT;    // 2*RBLK f
    float* mm      = partial + 2 * RBLK;      // 2 f
    float* out     = (float*)d_out;           // [16,224,224]

    const int total = NB * OUT * OUT;
    const int tb = 256;

    mahal_wmma_kernel<<<NP / MWAVES, 32 * MWAVES, 0, stream>>>(emb, means, invcov, dist);
    resize_bilinear_kernel<<<(total + tb - 1) / tb, tb, 0, stream>>>(dist, up);
    blur_h_kernel<<<(total + tb - 1) / tb, tb, 0, stream>>>(up, tmp);
    blur_w_kernel<<<(total + tb - 1) / tb, tb, 0, stream>>>(tmp, out);
    minmax_partial_kernel<<<RBLK, tb, 0, stream>>>(out, total, partial);
    minmax_final_kernel<<<1, tb, 0, stream>>>(partial, mm, RBLK);
    normalize_kernel<<<(total + tb - 1) / tb, tb, 0, stream>>>(out, mm, total);
}